// StaticSparseLinear_7275674599452
// MI455X (gfx1250) — compile-verified
//
#include <hip/hip_runtime.h>

typedef float v2f __attribute__((ext_vector_type(2)));
typedef float v4f __attribute__((ext_vector_type(4)));
typedef float v8f __attribute__((ext_vector_type(8)));

#define BLOCKS_OUT 128
#define BLOCKS_IN  128
#define BLOCK      64
#define NNZ        2048
#define BATCH      2048
#define XCOLS      (BLOCKS_IN * BLOCK)   /* 8192 */
#define OUTCOLS    (BLOCKS_OUT * BLOCK)  /* 8192 */

#define BT         128                   /* batch rows per workgroup */
#define XS_STRIDE  68                    /* x tile LDS row stride (floats): conflict-free frag reads */
#define BS_STRIDE  160                   /* valuesT kpair-row stride (floats): disjoint bank halves */

// ---------------------------------------------------------------------------
// Kernel 1: deterministic bucketing of nnz blocks by row.
//   offs[r]..offs[r+1] delimit bucket entries (nnz indices, ascending) for row r.
// ---------------------------------------------------------------------------
__global__ __launch_bounds__(256)
void ssl_bucket_kernel(const int* __restrict__ row_idx,
                       int* __restrict__ offs,     /* [BLOCKS_OUT+1] */
                       int* __restrict__ bucket)   /* [NNZ] */
{
    __shared__ int rows_s[NNZ];
    __shared__ int cnt_s[BLOCKS_OUT];
    __shared__ int off_s[BLOCKS_OUT + 1];
    const int tid = threadIdx.x;

    for (int n = tid; n < NNZ; n += 256) rows_s[n] = row_idx[n];
    if (tid < BLOCKS_OUT) cnt_s[tid] = 0;
    __syncthreads();

    for (int n = tid; n < NNZ; n += 256) atomicAdd(&cnt_s[rows_s[n]], 1);
    __syncthreads();

    if (tid == 0) {
        int run = 0;
        for (int r = 0; r < BLOCKS_OUT; ++r) { off_s[r] = run; run += cnt_s[r]; }
        off_s[BLOCKS_OUT] = run;
    }
    __syncthreads();

    if (tid <= BLOCKS_OUT) offs[tid] = off_s[tid];

    // Stable deterministic scatter: position = #prior entries with same row.
    for (int n = tid; n < NNZ; n += 256) {
        const int r = rows_s[n];
        int c = 0;
        for (int m = 0; m < n; ++m) c += (rows_s[m] == r) ? 1 : 0;
        bucket[off_s[r] + c] = n;
    }
}

// ---------------------------------------------------------------------------
// Kernel 2: block-COO SpMM via V_WMMA_F32_16X16X4_F32, register-double-buffered.
//   WG = (row_block, batch_tile of 128). 8 waves; wave w owns batch rows
//   [w*16, w*16+16) and the full 64 output columns (4 WMMA N-tiles).
//   C[m=batch][n=out] += A[m][k] * B[k][n], A = x tile, B = values^T.
//   Next iteration's global loads are issued before the compute phase so
//   they retire behind the 64-WMMA window.
// ---------------------------------------------------------------------------
__global__ __launch_bounds__(256)
void ssl_spmm_wmma_kernel(const float* __restrict__ x,
                          const float* __restrict__ values,
                          const int* __restrict__ col_idx,
                          const int* __restrict__ offs,
                          const int* __restrict__ bucket,
                          float* __restrict__ out)
{
    __shared__ float xs[BT * XS_STRIDE];   /* x tile  [128][68]          ~34.8 KB */
    __shared__ float bs[32 * BS_STRIDE];   /* B as [kpair=32][n=64][2]   ~20.5 KB */

    const int row  = blockIdx.x;
    const int bt   = blockIdx.y * BT;
    const int tid  = threadIdx.x;
    const int wave = tid >> 5;
    const int lane = tid & 31;
    const int half = lane >> 4;   /* lanes 16..31 hold K+2 (A) / M+8 (C)   */
    const int lm   = lane & 15;

    v8f acc[4] = {};               /* 4x 16x16 fp32 C tiles -> 32 VGPRs    */

    const int start = offs[row];
    const int end   = offs[row + 1];

    // Per-thread staging positions (fixed across iterations).
    // x tile: 2048 float4s over [128 rows][16 float4 cols]; 8 per thread.
    // values: 1024 float4s over 64x64; 4 per thread.
    v4f rx[8];
    v4f rv[4];

    // ---- prologue: prefetch first block's tiles into registers
    if (start < end) {
        const int nz0  = bucket[start];
        const int col0 = col_idx[nz0];
        const float* xsrc = x + (size_t)bt * XCOLS + (size_t)col0 * BLOCK;
        #pragma unroll
        for (int i = 0; i < 8; ++i) {
            const int linear = i * 256 + tid;
            const int r  = linear >> 4;
            const int c4 = (linear & 15) << 2;
            rx[i] = *(const v4f*)(xsrc + (size_t)r * XCOLS + c4);
        }
        const float* vsrc = values + (size_t)nz0 * (BLOCK * BLOCK);
        #pragma unroll
        for (int i4 = 0; i4 < 4; ++i4) {
            const int linear = i4 * 256 + tid;
            rv[i4] = *(const v4f*)(vsrc + (size_t)linear * 4);
        }
    }

    for (int it = start; it < end; ++it) {
        __syncthreads();   // previous iteration's compute done reading LDS

        // ---- commit prefetched tiles to LDS
        #pragma unroll
        for (int i = 0; i < 8; ++i) {
            const int linear = i * 256 + tid;
            const int r  = linear >> 4;
            const int c4 = (linear & 15) << 2;
            *(v4f*)&xs[r * XS_STRIDE + c4] = rx[i];   /* 68*4 % 16 == 0 -> aligned */
        }
        #pragma unroll
        for (int i4 = 0; i4 < 4; ++i4) {
            const int linear = i4 * 256 + tid;        /* float4 index over 64x64 */
            const int vi = linear >> 4;               /* i (out-feature) */
            const int vj = (linear & 15) << 2;        /* j (in-feature)  */
            #pragma unroll
            for (int p = 0; p < 4; ++p) {
                const int j = vj + p;
                bs[(j >> 1) * BS_STRIDE + vi * 2 + (j & 1)] = rv[i4][p];
            }
        }

        // ---- prefetch next block's tiles (retires behind the WMMA phase)
        if (it + 1 < end) {
            const int nz1  = bucket[it + 1];
            const int col1 = col_idx[nz1];
            const float* xsrc = x + (size_t)bt * XCOLS + (size_t)col1 * BLOCK;
            #pragma unroll
            for (int i = 0; i < 8; ++i) {
                const int linear = i * 256 + tid;
                const int r  = linear >> 4;
                const int c4 = (linear & 15) << 2;
                rx[i] = *(const v4f*)(xsrc + (size_t)r * XCOLS + c4);
            }
            const float* vsrc = values + (size_t)(bucket[it + 1]) * (BLOCK * BLOCK);
            #pragma unroll
            for (int i4 = 0; i4 < 4; ++i4) {
                const int linear = i4 * 256 + tid;
                rv[i4] = *(const v4f*)(vsrc + (size_t)linear * 4);
            }
        }

        __syncthreads();   // LDS tiles visible

        // ---- compute: 16 K-steps x 4 N-tiles of v_wmma_f32_16x16x4_f32
        const float* xw = &xs[(wave * 16 + lm) * XS_STRIDE];
        #pragma unroll
        for (int ks = 0; ks < 16; ++ks) {
            const int k0 = ks * 4;
            // A frag: VGPR0 = A[lm][k0+2*half], VGPR1 = A[lm][k0+2*half+1]
            v2f a = *(const v2f*)(xw + k0 + 2 * half);
            const float* brow = &bs[((k0 >> 1) + half) * BS_STRIDE];
            #pragma unroll
            for (int t = 0; t < 4; ++t) {
                v2f b = *(const v2f*)(brow + (t * 16 + lm) * 2);
                acc[t] = __builtin_amdgcn_wmma_f32_16x16x4_f32(
                    /*neg_a=*/false, a, /*neg_b=*/false, b,
                    /*c_mod=*/(short)0, acc[t],
                    /*reuse_a=*/false, /*reuse_b=*/false);
            }
        }
    }

    // ---- epilogue: C VGPR v, lane(half,lm) = C[m=v+8*half][n=lm]; streaming NT stores
    float* obase = out + (size_t)(bt + wave * 16) * OUTCOLS + (size_t)row * BLOCK;
    #pragma unroll
    for (int t = 0; t < 4; ++t) {
        #pragma unroll
        for (int v = 0; v < 8; ++v) {
            const int m = v + 8 * half;
            __builtin_nontemporal_store(acc[t][v],
                &obase[(size_t)m * OUTCOLS + t * 16 + lm]);
        }
    }
}

// ---------------------------------------------------------------------------
extern "C" void kernel_launch(void* const* d_in, const int* in_sizes, int n_in,
                              void* d_out, int out_size, void* d_ws, size_t ws_size,
                              hipStream_t stream) {
    const float* x       = (const float*)d_in[0];
    const float* values  = (const float*)d_in[1];
    const int*   row_idx = (const int*)d_in[2];
    const int*   col_idx = (const int*)d_in[3];
    float*       out     = (float*)d_out;

    int* offs   = (int*)d_ws;          /* [129] (+pad) */
    int* bucket = offs + 132;          /* [NNZ]        */

    hipLaunchKernelGGL(ssl_bucket_kernel, dim3(1), dim3(256), 0, stream,
                       row_idx, offs, bucket);

    dim3 grid(BLOCKS_OUT, BATCH / BT);
    hipLaunchKernelGGL(ssl_spmm_wmma_kernel, grid, dim3(256), 0, stream,
                       x, values, col_idx, offs, bucket, out);
}